// ContastiveLoss_47734266527855
// MI455X (gfx1250) — compile-verified
//
#include <hip/hip_runtime.h>
#include <math.h>

typedef float v2f __attribute__((ext_vector_type(2)));
typedef float v8f __attribute__((ext_vector_type(8)));

// ---------------- kernel 1: inverse row norms (torch-style clamp) ----------------
__global__ void k_rownorm(const float* __restrict__ x, float* __restrict__ invn, int D) {
  __shared__ float red[256];
  int row = blockIdx.x;
  const float* xr = x + (size_t)row * D;
  float s = 0.f;
  for (int i = threadIdx.x; i < D; i += 256) { float v = xr[i]; s += v * v; }
  red[threadIdx.x] = s;
  __syncthreads();
  for (int off = 128; off > 0; off >>= 1) {
    if ((int)threadIdx.x < off) red[threadIdx.x] += red[threadIdx.x + off];
    __syncthreads();
  }
  if (threadIdx.x == 0) invn[row] = 1.0f / fmaxf(sqrtf(red[0]), 1e-8f);
}

// ---------------- kernel 2: per-instance metadata + zero accumulators ----------------
__global__ void k_meta(const int* __restrict__ bag_idxs, const int* __restrict__ bag_labels,
                       const float* __restrict__ iscore, int* __restrict__ meta,
                       float* __restrict__ sE, float* __restrict__ sL, float* __restrict__ sM,
                       int N) {
  int i = blockIdx.x * 256 + threadIdx.x;
  if (i < N) {
    int bag   = bag_idxs[i];
    int label = bag_labels[bag];
    float sc  = iscore[i];
    int neg = ((label == 0) || (sc < 0.f)) ? 1 : 0;
    int hp  = ((sc > 1.0f) && (label == 1)) ? 1 : 0;
    meta[i] = (bag & 255) | (neg << 8) | (hp << 9);
    sE[i] = 0.f; sL[i] = 0.f; sM[i] = 0.f;
  }
}

// ---------------- kernel 3: fused fp32-WMMA cosine GEMM + masked row reductions ----------------
// One wave per workgroup. Wave computes a 64x64 block of logits over full K via
// V_WMMA_F32_16X16X4_F32 (16 accumulator tiles), then fuses exp / mask / row sums.
__global__ __launch_bounds__(32)
void k_gemm(const float* __restrict__ feat, const float* __restrict__ invn,
            const float* __restrict__ compass, const int* __restrict__ meta,
            float* __restrict__ sE, float* __restrict__ sL, float* __restrict__ sM,
            int D, int nCB) {
  const int lane = threadIdx.x;        // 0..31, wave32
  const int l15  = lane & 15;
  const int hi   = lane >> 4;          // 0: K=0,1  1: K=2,3 (A/B operand halves)
  const int rbase = (blockIdx.x / nCB) * 64;
  const int cbase = (blockIdx.x % nCB) * 64;

  v8f acc[4][4];
#pragma unroll
  for (int a = 0; a < 4; a++)
#pragma unroll
    for (int b = 0; b < 4; b++) acc[a][b] = (v8f){};

  const float* ap[4]; float sA[4];
  const float* bp[4]; float sB[4];
#pragma unroll
  for (int t = 0; t < 4; t++) {
    int r = rbase + t * 16 + l15;
    ap[t] = feat + (size_t)r * D + 2 * hi;
    sA[t] = invn[r];
    int c = cbase + t * 16 + l15;
    bp[t] = feat + (size_t)c * D + 2 * hi;
    sB[t] = invn[c];
  }

  // K loop: 8 float2 loads + 16 fp32 WMMAs per step; normalization folded into operands.
  for (int k = 0; k < D; k += 4) {
    v2f a[4], b[4];
#pragma unroll
    for (int t = 0; t < 4; t++) { a[t] = *(const v2f*)(ap[t] + k); a[t] *= sA[t]; }
#pragma unroll
    for (int t = 0; t < 4; t++) { b[t] = *(const v2f*)(bp[t] + k); b[t] *= sB[t]; }
#pragma unroll
    for (int ti = 0; ti < 4; ti++)
#pragma unroll
      for (int tj = 0; tj < 4; tj++)
        acc[ti][tj] = __builtin_amdgcn_wmma_f32_16x16x4_f32(
            false, a[ti], false, b[tj], (short)0, acc[ti][tj], false, false);
  }

  // ---- fused epilogue: exp / mask / per-row partial sums ----
  float ccx[4], ccy[4], ccz[4]; int mC[4];
#pragma unroll
  for (int tj = 0; tj < 4; tj++) {
    int c = cbase + tj * 16 + l15;
    ccx[tj] = compass[3 * c + 0];
    ccy[tj] = compass[3 * c + 1];
    ccz[tj] = compass[3 * c + 2];
    mC[tj]  = meta[c];
  }

#pragma unroll
  for (int ti = 0; ti < 4; ti++) {
#pragma unroll
    for (int v = 0; v < 8; v++) {
      int r = rbase + ti * 16 + v + 8 * hi;   // C layout: VGPR v -> row v (lanes 0-15) / v+8 (16-31)
      float crx = compass[3 * r + 0];
      float cry = compass[3 * r + 1];
      float crz = compass[3 * r + 2];
      int mR = meta[r];
      int bagR  = mR & 255;
      bool negR = (mR >> 8) & 1;
      bool hpR  = (mR >> 9) & 1;
      float e = 0.f, lm = 0.f, mc = 0.f;
#pragma unroll
      for (int tj = 0; tj < 4; tj++) {
        float logit = acc[ti][tj][v];
        int c = cbase + tj * 16 + l15;        // C layout: N = lane&15
        bool offd = (c != r);
        if (offd) e += __expf(logit);
        float dx = crx - ccx[tj], dy = cry - ccy[tj], dz = crz - ccz[tj];
        bool cok = (dx * dx + dy * dy + dz * dz) < 2500.0f;  // sqrt(d2) < 50
        int m = mC[tj];
        bool negp = negR && ((m >> 8) & 1);
        bool posp = ((m & 255) == bagR) && hpR && ((m >> 9) & 1);
        if (offd && cok && (negp || posp)) { lm += logit; mc += 1.f; }
      }
      // reduce across the 16 lanes striping this row's columns (xor<16 stays in half-wave)
#pragma unroll
      for (int s = 1; s < 16; s <<= 1) {
        e  += __shfl_xor(e,  s, 32);
        lm += __shfl_xor(lm, s, 32);
        mc += __shfl_xor(mc, s, 32);
      }
      if (l15 == 0) {  // lanes 0 and 16 write their own (different) rows
        atomicAdd(&sE[r], e);
        atomicAdd(&sL[r], lm);
        atomicAdd(&sM[r], mc);
      }
    }
  }
}

// ---------------- kernel 4: final scalar loss ----------------
__global__ void k_final(const float* __restrict__ sE, const float* __restrict__ sL,
                        const float* __restrict__ sM, float* __restrict__ out, int N) {
  __shared__ float red[256];
  float local = 0.f;
  for (int r = threadIdx.x; r < N; r += 256) {
    float denom = logf(sE[r]) + 1e-12f;
    float num = sL[r] - denom * sM[r];
    local += num / fmaxf(sM[r], 1e-8f);
  }
  red[threadIdx.x] = local;
  __syncthreads();
  for (int off = 128; off > 0; off >>= 1) {
    if ((int)threadIdx.x < off) red[threadIdx.x] += red[threadIdx.x + off];
    __syncthreads();
  }
  if (threadIdx.x == 0) out[0] = -red[0] / (float)N;
}

extern "C" void kernel_launch(void* const* d_in, const int* in_sizes, int n_in,
                              void* d_out, int out_size, void* d_ws, size_t ws_size,
                              hipStream_t stream) {
  const float* feat       = (const float*)d_in[0];
  const int*   bag_idxs   = (const int*)d_in[1];
  const int*   bag_labels = (const int*)d_in[2];
  const float* compass    = (const float*)d_in[3];
  const float* iscore     = (const float*)d_in[4];

  const int N = in_sizes[4];            // 4096
  const int D = in_sizes[0] / N;        // 1024

  // workspace layout: invn | meta | sumExp | sumLM | sumM  (~80 KB total)
  char* ws = (char*)d_ws;
  float* invn = (float*)(ws);
  int*   meta = (int*)  (ws + 1 * (size_t)N * 4);
  float* sE   = (float*)(ws + 2 * (size_t)N * 4);
  float* sL   = (float*)(ws + 3 * (size_t)N * 4);
  float* sM   = (float*)(ws + 4 * (size_t)N * 4);

  k_rownorm<<<N, 256, 0, stream>>>(feat, invn, D);
  k_meta<<<(N + 255) / 256, 256, 0, stream>>>(bag_idxs, bag_labels, iscore, meta, sE, sL, sM, N);

  const int nCB = N / 64;
  k_gemm<<<nCB * nCB, 32, 0, stream>>>(feat, invn, compass, meta, sE, sL, sM, D, nCB);

  k_final<<<1, 256, 0, stream>>>(sE, sL, sM, (float*)d_out, N);
}